// SoftKMeans_79800492359975
// MI455X (gfx1250) — compile-verified
//
#include <hip/hip_runtime.h>
#include <hip/hip_bf16.h>
#include <math.h>

// Problem constants (match reference)
#define NS 256   // n_samples
#define KC 64    // n_clusters
#define TT 128   // seq_len
#define FF 64    // n_features
#define ALPHA 1.0f

typedef __attribute__((ext_vector_type(2))) float v2f;
typedef __attribute__((ext_vector_type(8))) float v8f;

// Byte offset of a generic pointer within LDS (for async-to-LDS VDST operand).
__device__ __forceinline__ unsigned lds_off(const void* p) {
    return (unsigned)(uintptr_t)(__attribute__((address_space(3))) const void*)p;
}

// ---------------------------------------------------------------------------
// Kernel 0: x2[n,f] = sum_t X[n,t,f]^2 ; c2[k,f] = sum_t C[k,t,f]^2
// ---------------------------------------------------------------------------
__global__ __launch_bounds__(256) void k_sq(const float* __restrict__ X,
                                            const float* __restrict__ Cm,
                                            float* __restrict__ x2,
                                            float* __restrict__ c2) {
    int idx = blockIdx.x * blockDim.x + threadIdx.x;
    if (idx < NS * FF) {
        int n = idx / FF, f = idx % FF;
        const float* p = X + (size_t)n * TT * FF + f;
        float s = 0.0f;
        for (int t = 0; t < TT; ++t) { float v = p[t * FF]; s += v * v; }
        x2[idx] = s;
    } else if (idx < NS * FF + KC * FF) {
        int j = idx - NS * FF;
        int k = j / FF, f = j % FF;
        const float* p = Cm + (size_t)k * TT * FF + f;
        float s = 0.0f;
        for (int t = 0; t < TT; ++t) { float v = p[t * FF]; s += v * v; }
        c2[j] = s;
    }
}

// ---------------------------------------------------------------------------
// Kernel 1: distance[n,k] = sum_f sqrt(max(x2 + c2 - 2*xc, 0))
// xc computed with V_WMMA_F32_16X16X4_F32, contraction over T (32 steps).
// Block: 128 threads = 4 waves; wave w owns features [16w, 16w+16).
// Grid: (N/16, K/16) = (16, 4) tiles of 16x16.
// ---------------------------------------------------------------------------
__global__ __launch_bounds__(128) void k_distance(const float* __restrict__ X,
                                                  const float* __restrict__ Cm,
                                                  const float* __restrict__ x2,
                                                  const float* __restrict__ c2,
                                                  float* __restrict__ dist) {
    __shared__ float sdist[256];
    const int tid  = threadIdx.x;
    const int lane = tid & 31;
    const int wave = tid >> 5;          // 0..3
    const int nb   = blockIdx.x * 16;
    const int kb   = blockIdx.y * 16;

    sdist[tid] = 0.0f;
    sdist[tid + 128] = 0.0f;
    __syncthreads();

    const int m  = lane & 15;           // A row / B column index within tile
    const int hi = lane >> 4;           // half-wave select
    const int kk = hi * 2;              // K-slot offset inside 16x16x4 step

    const float* xrow = X  + (size_t)(nb + m) * TT * FF;
    const float* crow = Cm + (size_t)(kb + m) * TT * FF;

    v8f dacc = 0.0f;                    // per-lane partial distances (8 rows)

    for (int fi = 0; fi < 16; ++fi) {
        const int f = wave * 16 + fi;
        v8f acc = 0.0f;
        for (int t = 0; t < TT; t += 4) {
            v2f a, b;
            // A 16x4 f32 layout: lanes0-15 hold K={t,t+1}, lanes16-31 K={t+2,t+3}
            a[0] = xrow[(t + kk    ) * FF + f];
            a[1] = xrow[(t + kk + 1) * FF + f];
            b[0] = crow[(t + kk    ) * FF + f];
            b[1] = crow[(t + kk + 1) * FF + f];
            acc = __builtin_amdgcn_wmma_f32_16x16x4_f32(
                false, a, false, b, (short)0, acc, false, false);
        }
        // C/D layout: VGPR r -> M = r + 8*hi, N = m
        const float c2v = c2[(kb + m) * FF + f];
        #pragma unroll
        for (int r = 0; r < 8; ++r) {
            const int n = nb + r + 8 * hi;
            float d2 = x2[n * FF + f] + c2v - 2.0f * acc[r];
            d2 = fmaxf(d2, 0.0f);
            dacc[r] += sqrtf(d2);
        }
    }

    // Merge the 4 waves' feature-partials via LDS float atomics (ds_add_f32).
    #pragma unroll
    for (int r = 0; r < 8; ++r) {
        atomicAdd(&sdist[(r + 8 * hi) * 16 + m], dacc[r]);
    }
    __syncthreads();

    for (int e = tid; e < 256; e += 128) {
        int rn = e >> 4, rk = e & 15;
        dist[(nb + rn) * KC + (kb + rk)] = sdist[e];
    }
}

// ---------------------------------------------------------------------------
// Kernel 2 (fused tail): stage dist into LDS with async-to-LDS copies, then
//   q  = softmax_k(1/(1+d^2))              (overwrites LDS tile, also -> out)
//   p  = softmax_k(q^2/colsum)             (colsum recomputed per wave)
//   loss = sum(q*(log q - p)) / N          (deterministic tree reduction)
// One block, 256 threads = 8 waves; wave w owns rows [32w, 32w+32).
// LDS row-major n*64+lane => 32 lanes hit 32 distinct banks everywhere.
// ---------------------------------------------------------------------------
__global__ __launch_bounds__(256) void k_post(const float* __restrict__ dist,
                                              float* __restrict__ qout,
                                              float* __restrict__ lossp) {
    __shared__ float sq[NS * KC];       // 64 KB
    const int tid  = threadIdx.x;
    const int lane = tid & 31;
    const int wave = tid >> 5;          // 0..7

    // ---- Phase A: dist (N*K floats) global -> LDS via async B128 copies ----
    {
        const unsigned lbase = lds_off(&sq[0]);
        #pragma unroll
        for (int i = 0; i < (NS * KC / 4) / 256; ++i) {   // 16 iterations
            const int g = tid + 256 * i;                   // float4 index
            const unsigned loff = lbase + (unsigned)g * 16u;
            const unsigned long long gaddr =
                (unsigned long long)(uintptr_t)(dist + (size_t)g * 4);
            asm volatile("global_load_async_to_lds_b128 %0, %1, off"
                         :: "v"(loff), "v"(gaddr) : "memory");
        }
        asm volatile("s_wait_asynccnt 0x0" ::: "memory");
    }
    __syncthreads();

    // ---- Phase B: q = row softmax of 1/(1+d^2); overwrite LDS, emit q ----
    for (int r = 0; r < 32; ++r) {
        const int n = wave * 32 + r;
        float d0 = sq[n * KC + lane];
        float d1 = sq[n * KC + 32 + lane];
        float u0 = 1.0f / (1.0f + d0 * d0 / ALPHA);   // ^((alpha+1)/2) == id
        float u1 = 1.0f / (1.0f + d1 * d1 / ALPHA);
        float mx = fmaxf(u0, u1);
        for (int off = 16; off; off >>= 1) mx = fmaxf(mx, __shfl_xor(mx, off, 32));
        float e0 = expf(u0 - mx), e1 = expf(u1 - mx);
        float s = e0 + e1;
        for (int off = 16; off; off >>= 1) s += __shfl_xor(s, off, 32);
        float inv = 1.0f / s;
        float q0 = e0 * inv, q1 = e1 * inv;
        sq[n * KC + lane]      = q0;
        sq[n * KC + 32 + lane] = q1;
        qout[n * KC + lane]      = q0;
        qout[n * KC + 32 + lane] = q1;
    }
    __syncthreads();

    // ---- Phase C: per-wave column sums (each wave computes all 64) ----
    float cs0 = 0.0f, cs1 = 0.0f;
    for (int n = 0; n < NS; ++n) {
        cs0 += sq[n * KC + lane];
        cs1 += sq[n * KC + 32 + lane];
    }
    float inv_cs0 = 1.0f / cs0, inv_cs1 = 1.0f / cs1;

    // ---- Phase D: p softmax + per-wave loss accumulation ----
    float wl = 0.0f;
    for (int r = 0; r < 32; ++r) {
        const int n = wave * 32 + r;
        float q0 = sq[n * KC + lane];
        float q1 = sq[n * KC + 32 + lane];
        float s0 = q0 * q0 * inv_cs0;
        float s1 = q1 * q1 * inv_cs1;
        float mx = fmaxf(s0, s1);
        for (int off = 16; off; off >>= 1) mx = fmaxf(mx, __shfl_xor(mx, off, 32));
        float e0 = expf(s0 - mx), e1 = expf(s1 - mx);
        float s = e0 + e1;
        for (int off = 16; off; off >>= 1) s += __shfl_xor(s, off, 32);
        float inv = 1.0f / s;
        float p0 = e0 * inv, p1 = e1 * inv;
        float l = q0 * (logf(q0) - p0) + q1 * (logf(q1) - p1);
        for (int off = 16; off; off >>= 1) l += __shfl_xor(l, off, 32);
        wl += l;                         // identical in all lanes of the wave
    }
    __syncthreads();                     // all LDS q reads complete
    if (lane == 0) sq[wave] = wl;        // reuse LDS for 8 wave partials
    __syncthreads();
    if (tid == 0) {
        float s = 0.0f;
        for (int w = 0; w < 8; ++w) s += sq[w];
        *lossp = s / (float)NS;
    }
}

// ---------------------------------------------------------------------------
// Workspace layout (floats):
//   [0)          x2      N*F = 16384
//   [16384)      c2      K*F =  4096
//   [20480)      dist    N*K = 16384
// Output: d_out[0..16383] = q (N,K);  d_out[16384] = loss.
// ---------------------------------------------------------------------------
extern "C" void kernel_launch(void* const* d_in, const int* in_sizes, int n_in,
                              void* d_out, int out_size, void* d_ws, size_t ws_size,
                              hipStream_t stream) {
    (void)in_sizes; (void)n_in; (void)out_size; (void)ws_size;

    const float* X  = (const float*)d_in[0];   // (N, T, F)
    const float* Cm = (const float*)d_in[1];   // (K, T, F)
    float* out = (float*)d_out;

    float* ws   = (float*)d_ws;
    float* x2   = ws;
    float* c2   = ws + 16384;
    float* dist = ws + 20480;

    float* qout  = out;                 // (N, K)
    float* lossp = out + NS * KC;

    // 0) squared row norms
    {
        int total = NS * FF + KC * FF;  // 20480
        k_sq<<<(total + 255) / 256, 256, 0, stream>>>(X, Cm, x2, c2);
    }
    // 1) WMMA batched contraction + distance
    {
        dim3 grid(NS / 16, KC / 16);    // (16, 4)
        k_distance<<<grid, 128, 0, stream>>>(X, Cm, x2, c2, dist);
    }
    // 2) fused softmax / target / loss tail (async-to-LDS staging)
    k_post<<<1, 256, 0, stream>>>(dist, qout, lossp);
}